// MultiLayerGNN_1864015807061
// MI455X (gfx1250) — compile-verified
//
#include <hip/hip_runtime.h>

#define DFEAT 256
#define ASTRIDE 260   // padded LDS row stride (floats): rows land on different banks
#define LN_EPS 1e-5f

typedef __attribute__((ext_vector_type(2))) float v2f;
typedef __attribute__((ext_vector_type(8))) float v8f;

// ---------------------------------------------------------------------------
// GEMM: out = h @ W   (h: [N,256] f32 row-major, W: [256,256] f32 row-major)
// Block = 128 threads (4 waves). Block computes 32 rows x 256 cols.
// Each wave owns a 64-col strip and BOTH 16-row tiles: per K-step (K=4) it
// issues 2 ds_load_2addr_b64 (A frags) + 8 global_load_b32 (B frags) and
// feeds 8 V_WMMA_F32_16X16X4_F32 ops -> B fragments reused x2 in registers,
// halving W read traffic vs a 16-row block.
// All 16-row tiles are full (N % 16 == 0); the possibly-missing second tile
// of the last block is handled with clamped staging reads + a block-uniform
// store guard, so EXEC is all-ones through every WMMA.
// ---------------------------------------------------------------------------
__global__ __launch_bounds__(128) void gcn_gemm_f32(
    const float* __restrict__ h, const float* __restrict__ W,
    float* __restrict__ out, int N)
{
  __shared__ float As[32 * ASTRIDE];
  const int row0 = blockIdx.x * 32;

  // Stage the 32x256 A tile into LDS (coalesced float4 loads, rows clamped).
  {
    const int t = threadIdx.x;
#pragma unroll
    for (int i = 0; i < 16; ++i) {
      int idx = t + 128 * i;        // 2048 float4 slots = 32 rows * 64
      int r   = idx >> 6;
      int c4  = idx & 63;
      int rg  = row0 + r;
      if (rg > N - 1) rg = N - 1;   // clamp: valid read, result discarded
      float4 v = *reinterpret_cast<const float4*>(
          h + (size_t)rg * DFEAT + c4 * 4);
      float* dst = &As[r * ASTRIDE + c4 * 4];
      dst[0] = v.x; dst[1] = v.y; dst[2] = v.z; dst[3] = v.w;
    }
  }
  __syncthreads();

  const int lane = threadIdx.x & 31;
  const int wave = threadIdx.x >> 5;
  const int nIdx = lane & 15;   // N (and M) position within half-wave
  const int kh   = lane >> 4;   // 0 -> holds K0/K1, 1 -> holds K2/K3
  const int col0 = wave * 64;

  v8f acc[2][4];
#pragma unroll
  for (int rt = 0; rt < 2; ++rt)
#pragma unroll
    for (int ct = 0; ct < 4; ++ct) acc[rt][ct] = (v8f){};

#pragma unroll 4
  for (int k = 0; k < DFEAT; k += 4) {
    // A fragments (16x4): lanes 0-15 = M rows with K pair 0/1, lanes 16-31 K 2/3
    const float* ap0 = &As[nIdx * ASTRIDE + k + 2 * kh];
    const float* ap1 = ap0 + 16 * ASTRIDE;
    v2f a0; a0.x = ap0[0]; a0.y = ap0[1];
    v2f a1; a1.x = ap1[0]; a1.y = ap1[1];

    // B fragments (4x16): element B[kk][n] = W[kk][col0 + ct*16 + n]
    const float* wp = W + (size_t)(k + 2 * kh) * DFEAT + col0 + nIdx;
    v2f b[4];
#pragma unroll
    for (int ct = 0; ct < 4; ++ct) {
      b[ct].x = wp[16 * ct];
      b[ct].y = wp[DFEAT + 16 * ct];
    }

#pragma unroll
    for (int ct = 0; ct < 4; ++ct) {
      acc[0][ct] = __builtin_amdgcn_wmma_f32_16x16x4_f32(
          false, a0, false, b[ct], (short)0, acc[0][ct], false, false);
      acc[1][ct] = __builtin_amdgcn_wmma_f32_16x16x4_f32(
          false, a1, false, b[ct], (short)0, acc[1][ct], false, false);
    }
  }

  // C/D layout: VGPR v -> M=v (lanes 0-15) / M=v+8 (lanes 16-31); N = lane&15
#pragma unroll
  for (int rt = 0; rt < 2; ++rt) {
    const int rbase = row0 + rt * 16;
    if (rbase >= N) break;          // block-uniform: tail block second tile
#pragma unroll
    for (int v = 0; v < 8; ++v) {
      size_t o = (size_t)(rbase + v + 8 * kh) * DFEAT + col0 + nIdx;
      out[o +  0] = acc[rt][0][v];
      out[o + 16] = acc[rt][1][v];
      out[o + 32] = acc[rt][2][v];
      out[o + 48] = acc[rt][3][v];
    }
  }
}

// ---------------------------------------------------------------------------
// row is sorted -> CSR row_ptr via per-node lower_bound binary search.
// ---------------------------------------------------------------------------
__global__ void build_rowptr(const int* __restrict__ row,
                             int* __restrict__ rowptr, int N, int E)
{
  int n = blockIdx.x * blockDim.x + threadIdx.x;
  if (n > N) return;
  int lo = 0, hi = E;
  while (lo < hi) {
    int mid = (lo + hi) >> 1;
    if (row[mid] < n) lo = mid + 1; else hi = mid;
  }
  rowptr[n] = lo;
}

// ---------------------------------------------------------------------------
// Fused: agg = segsum(vals * support[col]) ; out = agg + b ; (relu) ;
//        out += h_in ; h_out = LayerNorm(out) * g + beta
// One block (256 threads, 8 waves) per node; thread j = feature j.
// Gather rows of `support` are L2-resident (51 MB < 192 MB L2); next edge's
// row is prefetched (global_prefetch_b8) to hide L2 latency behind the FMA.
// ---------------------------------------------------------------------------
__global__ __launch_bounds__(256) void gcn_agg_ln(
    const float* __restrict__ support, const float* __restrict__ hin,
    const float* __restrict__ vals,    const int* __restrict__ col,
    const int* __restrict__ rowptr,    const float* __restrict__ bias,
    const float* __restrict__ gamma,   const float* __restrict__ beta,
    float* __restrict__ hout, int do_relu)
{
  const int n = blockIdx.x;
  const int j = threadIdx.x;

  const int e0 = rowptr[n];
  const int e1 = rowptr[n + 1];

  float acc = 0.0f;
  for (int e = e0; e < e1; ++e) {
    if (e + 1 < e1)
      __builtin_prefetch(&support[(size_t)col[e + 1] * DFEAT + j], 0, 1);
    const float v = vals[e];      // uniform across block -> scalarized
    const int   c = col[e];
    acc = fmaf(v, support[(size_t)c * DFEAT + j], acc);
  }

  float out = acc + bias[j];
  if (do_relu) out = fmaxf(out, 0.0f);
  out += hin[(size_t)n * DFEAT + j];

  // LayerNorm across 256 features: wave32 shuffle reduce + LDS combine
  float s = out, s2 = out * out;
#pragma unroll
  for (int off = 16; off; off >>= 1) {
    s  += __shfl_xor(s,  off, 32);
    s2 += __shfl_xor(s2, off, 32);
  }

  __shared__ float part[16];
  __shared__ float stat[2];
  const int w = j >> 5, l = j & 31;
  if (l == 0) { part[w] = s; part[8 + w] = s2; }
  __syncthreads();
  if (j == 0) {
    float S = 0.0f, S2 = 0.0f;
#pragma unroll
    for (int i = 0; i < 8; ++i) { S += part[i]; S2 += part[8 + i]; }
    const float mu  = S * (1.0f / DFEAT);
    const float var = S2 * (1.0f / DFEAT) - mu * mu;
    stat[0] = mu;
    stat[1] = rsqrtf(var + LN_EPS);
  }
  __syncthreads();

  const float mu = stat[0], rinv = stat[1];
  hout[(size_t)n * DFEAT + j] = (out - mu) * rinv * gamma[j] + beta[j];
}

// ---------------------------------------------------------------------------
extern "C" void kernel_launch(void* const* d_in, const int* in_sizes, int n_in,
                              void* d_out, int out_size, void* d_ws, size_t ws_size,
                              hipStream_t stream)
{
  const float* x     = (const float*)d_in[0];  // [N,256]
  const float* vals  = (const float*)d_in[1];  // [E]
  const float* Ws    = (const float*)d_in[2];  // [3,256,256]
  const float* bs    = (const float*)d_in[3];  // [3,256]
  const float* gs    = (const float*)d_in[4];  // [3,256]
  const float* betas = (const float*)d_in[5];  // [3,256]
  const int*   row   = (const int*)d_in[6];    // [E] sorted
  const int*   col   = (const int*)d_in[7];    // [E]

  const int N = in_sizes[0] / DFEAT;
  const int E = in_sizes[1];

  float* out = (float*)d_out;                       // [N,256], also h ping-pong
  char*  ws  = (char*)d_ws;
  float* support = (float*)ws;                      // N*256 f32
  int*   rowptr  = (int*)(ws + (size_t)N * DFEAT * sizeof(float)); // N+1 ints

  build_rowptr<<<(N + 1 + 255) / 256, 256, 0, stream>>>(row, rowptr, N, E);

  const int gemm_blocks = (N + 31) / 32;   // 32 rows per block

  // Layer 0: x -> out
  gcn_gemm_f32<<<gemm_blocks, 128, 0, stream>>>(x, Ws + 0 * DFEAT * DFEAT, support, N);
  gcn_agg_ln<<<N, 256, 0, stream>>>(support, x, vals, col, rowptr,
                                    bs + 0 * DFEAT, gs + 0 * DFEAT, betas + 0 * DFEAT,
                                    out, /*relu=*/1);
  // Layer 1: out -> out (element-wise in n, in-place safe)
  gcn_gemm_f32<<<gemm_blocks, 128, 0, stream>>>(out, Ws + 1 * DFEAT * DFEAT, support, N);
  gcn_agg_ln<<<N, 256, 0, stream>>>(support, out, vals, col, rowptr,
                                    bs + 1 * DFEAT, gs + 1 * DFEAT, betas + 1 * DFEAT,
                                    out, /*relu=*/1);
  // Layer 2: out -> out, no relu
  gcn_gemm_f32<<<gemm_blocks, 128, 0, stream>>>(out, Ws + 2 * DFEAT * DFEAT, support, N);
  gcn_agg_ln<<<N, 256, 0, stream>>>(support, out, vals, col, rowptr,
                                    bs + 2 * DFEAT, gs + 2 * DFEAT, betas + 2 * DFEAT,
                                    out, /*relu=*/0);
}